// GaussianMVNLL_86612310491494
// MI455X (gfx1250) — compile-verified
//
#include <hip/hip_runtime.h>
#include <math.h>

// GaussianMVNLL on MI455X (gfx1250): per-batch SYRK via V_WMMA_F32_16X16X4_F32,
// in-LDS Cholesky (2-wide packed update), register forward-solve, deterministic mean.

typedef __attribute__((ext_vector_type(2))) float v2f;
typedef __attribute__((ext_vector_type(8))) float v8f;

#define DIM   64
#define LDG   66          // EVEN padded LDS row stride: uniform 8B alignment for the
                          // 2-wide k-loop; (2i+j)%64 keeps column walks conflict-free
#define EPSV  1e-6f

__device__ __forceinline__ void store_tile(float* __restrict__ Gs, const v8f a,
                                           int ti, int tj, int ln, int h) {
  // C/D layout: lane = n + 16h, VGPR v holds (m = v + 8h, n)
#pragma unroll
  for (int v = 0; v < 8; ++v)
    Gs[(ti * 16 + v + 8 * h) * LDG + (tj * 16 + ln)] = a[v];
}

__global__ __launch_bounds__(32) void mvnll_batch_kernel(
    const float* __restrict__ means,
    const float* __restrict__ targets,
    const float* __restrict__ L,
    float* __restrict__ partial)
{
  __shared__ __align__(16) float Gs[DIM * LDG];  // Gram / Cholesky factor (lower valid)
  __shared__ __align__(8)  float colj[DIM];      // contiguous scaled pivot column

  const int b    = blockIdx.x;
  const int lane = threadIdx.x;     // wave32: 0..31
  const int ln   = lane & 15;       // row index within a 16-tile
  const int h    = lane >> 4;       // half selects K pair {2h, 2h+1}

  const float* __restrict__ Lb = L + (size_t)b * (DIM * DIM);

  // ---------------- SYRK: G = L * L^T, lower-triangular 16x16 tiles ----------------
  v8f a00 = {}, a10 = {}, a11 = {}, a20 = {}, a21 = {}, a22 = {},
      a30 = {}, a31 = {}, a32 = {}, a33 = {};

#pragma unroll 4
  for (int s = 0; s < 16; ++s) {
    const int c = 4 * s + 2 * h;   // K columns for this lane-half
    // A-frag(t) == B-frag(t): lane ln+16h holds (L[t*16+ln][c], L[t*16+ln][c+1])
    v2f f0 = *(const v2f*)(Lb + (0 * 16 + ln) * DIM + c);
    v2f f1 = *(const v2f*)(Lb + (1 * 16 + ln) * DIM + c);
    v2f f2 = *(const v2f*)(Lb + (2 * 16 + ln) * DIM + c);
    v2f f3 = *(const v2f*)(Lb + (3 * 16 + ln) * DIM + c);

    a00 = __builtin_amdgcn_wmma_f32_16x16x4_f32(false, f0, false, f0, (short)0, a00, false, false);
    a10 = __builtin_amdgcn_wmma_f32_16x16x4_f32(false, f1, false, f0, (short)0, a10, false, false);
    a11 = __builtin_amdgcn_wmma_f32_16x16x4_f32(false, f1, false, f1, (short)0, a11, false, false);
    a20 = __builtin_amdgcn_wmma_f32_16x16x4_f32(false, f2, false, f0, (short)0, a20, false, false);
    a21 = __builtin_amdgcn_wmma_f32_16x16x4_f32(false, f2, false, f1, (short)0, a21, false, false);
    a22 = __builtin_amdgcn_wmma_f32_16x16x4_f32(false, f2, false, f2, (short)0, a22, false, false);
    a30 = __builtin_amdgcn_wmma_f32_16x16x4_f32(false, f3, false, f0, (short)0, a30, false, false);
    a31 = __builtin_amdgcn_wmma_f32_16x16x4_f32(false, f3, false, f1, (short)0, a31, false, false);
    a32 = __builtin_amdgcn_wmma_f32_16x16x4_f32(false, f3, false, f2, (short)0, a32, false, false);
    a33 = __builtin_amdgcn_wmma_f32_16x16x4_f32(false, f3, false, f3, (short)0, a33, false, false);
  }

  store_tile(Gs, a00, 0, 0, ln, h);
  store_tile(Gs, a10, 1, 0, ln, h);
  store_tile(Gs, a11, 1, 1, ln, h);
  store_tile(Gs, a20, 2, 0, ln, h);
  store_tile(Gs, a21, 2, 1, ln, h);
  store_tile(Gs, a22, 2, 2, ln, h);
  store_tile(Gs, a30, 3, 0, ln, h);
  store_tile(Gs, a31, 3, 1, ln, h);
  store_tile(Gs, a32, 3, 2, ln, h);
  store_tile(Gs, a33, 3, 3, ln, h);
  __syncthreads();

  // cov = G + eps*I
  for (int j = lane; j < DIM; j += 32) Gs[j * LDG + j] += EPSV;
  __syncthreads();

  // ---------------- In-place Cholesky (lower triangle) ----------------
  for (int j = 0; j < DIM; ++j) {
    const float djj = sqrtf(Gs[j * LDG + j]);   // uniform across the wave
    const float inv = 1.0f / djj;
    if (lane == 0) Gs[j * LDG + j] = djj;
    // scale pivot column, mirror into contiguous colj[] (broadcast-friendly)
    for (int i = j + 1 + lane; i < DIM; i += 32) {
      const float v = Gs[i * LDG + j] * inv;
      Gs[i * LDG + j] = v;
      colj[i] = v;
    }
    __syncthreads();
    // rank-1 trailing update, two k's per step:
    // LDG is even and k is head-aligned to even -> every b64 LDS access is 8B-aligned
    for (int i = j + 1 + lane; i < DIM; i += 32) {
      const float lij = colj[i];
      float* __restrict__ row = Gs + i * LDG;
      int k = j + 1;
      if (k & 1) {                       // scalar head to reach even k
        row[k] -= lij * colj[k];
        ++k;
      }
#pragma unroll 2
      for (; k + 1 <= i; k += 2) {       // ds_load_b64 + packed fma + ds_store_b64
        v2f g = *(v2f*)(row + k);
        v2f cc = *(const v2f*)(colj + k);
        g -= lij * cc;
        *(v2f*)(row + k) = g;
      }
      if (k <= i)                        // scalar tail (per-lane parity of i)
        row[k] -= lij * colj[k];
    }
    __syncthreads();
  }

  // ---------------- Forward solve C y = diff, logdet, quad ----------------
  float r0 = targets[(size_t)b * DIM + lane]      - means[(size_t)b * DIM + lane];
  float r1 = targets[(size_t)b * DIM + lane + 32] - means[(size_t)b * DIM + lane + 32];
  float sumlog = 0.0f, quad = 0.0f;

  for (int j = 0; j < DIM; ++j) {
    const float rj  = (j < 32) ? __shfl(r0, j, 32) : __shfl(r1, j - 32, 32);
    const float cjj = Gs[j * LDG + j];            // LDS broadcast, uniform
    const float yj  = rj / cjj;
    sumlog += __logf(cjj);
    quad   += yj * yj;
    if (lane > j)      r0 -= Gs[lane * LDG + j] * yj;
    if (lane + 32 > j) r1 -= Gs[(lane + 32) * LDG + j] * yj;
  }

  // 0.5*(logdet + quad) = sum(log cjj) + 0.5*quad   (uniform across wave)
  if (lane == 0) partial[b] = sumlog + 0.5f * quad;
}

__global__ __launch_bounds__(256) void reduce_mean_kernel(
    const float* __restrict__ partial, float* __restrict__ out, int n)
{
  __shared__ float s[256];
  float acc = 0.0f;
  for (int i = threadIdx.x; i < n; i += 256) acc += partial[i];
  s[threadIdx.x] = acc;
  __syncthreads();
  for (int w = 128; w > 0; w >>= 1) {
    if ((int)threadIdx.x < w) s[threadIdx.x] += s[threadIdx.x + w];
    __syncthreads();
  }
  if (threadIdx.x == 0) out[0] = s[0] / (float)n;
}

extern "C" void kernel_launch(void* const* d_in, const int* in_sizes, int n_in,
                              void* d_out, int out_size, void* d_ws, size_t ws_size,
                              hipStream_t stream) {
  const float* means   = (const float*)d_in[0];
  const float* targets = (const float*)d_in[1];
  const float* L       = (const float*)d_in[2];
  float* out     = (float*)d_out;
  float* partial = (float*)d_ws;                 // B floats of scratch

  const int B = in_sizes[0] / DIM;               // 8192

  mvnll_batch_kernel<<<B, 32, 0, stream>>>(means, targets, L, partial);
  reduce_mean_kernel<<<1, 256, 0, stream>>>(partial, out, B);
}